// ShapeTimeDistortionLoss_24739011625011
// MI455X (gfx1250) — compile-verified
//
#include <hip/hip_runtime.h>
#include <hip/hip_bf16.h>
#include <math.h>

// DILATE loss on MI455X (gfx1250, wave32).
// Pipeline: prep (shift + norms) -> fp32 WMMA batched GEMM for the cost
// matrix -> soft-DTW forward wavefront (LDS diagonals) -> soft-DTW backward
// wavefront with fused E*(i-j)^2 reduction -> scalar finalize.

#define NBATCH 32
#define NLEN   512
#define NDIM   16
#define NPAD   513            // (N+1) padded DP grid rows/cols 0..512
#define BIGV   1e10f          // matches reference border stand-in
#define GAMMA_C   0.001f
#define INV_GAMMA 1000.0f

typedef float v2f __attribute__((ext_vector_type(2)));
typedef float v8f __attribute__((ext_vector_type(8)));

// ---------------------------------------------------------------- prep ----
// x = inp - inp[:,0,:]; tn[i] = ||tgt_i||^2 ; xn[j] = ||x_j||^2
__global__ __launch_bounds__(256) void prep_kernel(const float* __restrict__ inp,
                                                   const float* __restrict__ tgt,
                                                   float* __restrict__ xb,
                                                   float* __restrict__ tn,
                                                   float* __restrict__ xn) {
  const int r = blockIdx.x * 256 + threadIdx.x;        // 0 .. B*N-1
  if (r >= NBATCH * NLEN) return;
  const int b = r >> 9;
  const float* inRow = inp + (size_t)r * NDIM;
  const float* in0   = inp + (size_t)b * NLEN * NDIM;  // first step of batch
  const float* tgRow = tgt + (size_t)r * NDIM;
  float xacc = 0.0f, tacc = 0.0f;
#pragma unroll
  for (int k = 0; k < NDIM; ++k) {
    const float xv = inRow[k] - in0[k];
    xb[(size_t)r * NDIM + k] = xv;
    xacc += xv * xv;
    const float tv = tgRow[k];
    tacc += tv * tv;
  }
  xn[r] = xacc;
  tn[r] = tacc;
}

// ------------------------------------------------------- cost matrix ------
// D[b,i,j] = max(0, tn[i] + xn[j] - 2 * (tgt_i . x_j))  via fp32 WMMA.
// One wave computes one 16x16 tile of G = tgt @ x^T, K=16 as 4 x (16x16x4).
// Layouts per CDNA5 ISA 7.12.2:
//   A (16x4 f32): lanes 0-15 hold M=lane, VGPR0/1 = K{0,1}; lanes 16-31 K{2,3}
//   B (4x16 f32): mirrored over N
//   C/D (16x16 f32): acc[r] = G[M = r + 8*(lane>=16)][N = lane&15]
__global__ __launch_bounds__(256) void dist_wmma_kernel(const float* __restrict__ tgt,
                                                        const float* __restrict__ xb,
                                                        const float* __restrict__ tn,
                                                        const float* __restrict__ xn,
                                                        float* __restrict__ Dm) {
  const int wave = (blockIdx.x * 256 + (int)threadIdx.x) >> 5;  // wave32
  const int lane = threadIdx.x & 31;
  const int tilesPerRow   = NLEN / 16;               // 32
  const int tilesPerBatch = tilesPerRow * tilesPerRow; // 1024
  const int b  = wave / tilesPerBatch;
  const int t  = wave - b * tilesPerBatch;
  const int ti = t / tilesPerRow;
  const int tj = t - ti * tilesPerRow;
  const int l16  = lane & 15;
  const int half = lane >> 4;

  const float* Arow = tgt + ((size_t)b * NLEN + ti * 16 + l16) * NDIM; // tgt row
  const float* Brow = xb  + ((size_t)b * NLEN + tj * 16 + l16) * NDIM; // x row (col of B)

  v8f acc = {};
#pragma unroll
  for (int kb = 0; kb < NDIM; kb += 4) {
    v2f av, bv;
    av.x = Arow[kb + half * 2 + 0];
    av.y = Arow[kb + half * 2 + 1];
    bv.x = Brow[kb + half * 2 + 0];
    bv.y = Brow[kb + half * 2 + 1];
    // (neg_a, A, neg_b, B, c_mod, C, reuse_a, reuse_b)
    acc = __builtin_amdgcn_wmma_f32_16x16x4_f32(false, av, false, bv,
                                                (short)0, acc, false, false);
  }

  const int j = tj * 16 + l16;
  const float xnj = xn[b * NLEN + j];
  float* Drow = Dm + (size_t)b * NLEN * NLEN;
#pragma unroll
  for (int r = 0; r < 8; ++r) {
    const int icell = ti * 16 + r + half * 8;
    const float dv = tn[b * NLEN + icell] + xnj - 2.0f * acc[r];
    Drow[(size_t)icell * NLEN + j] = fmaxf(dv, 0.0f);
  }
}

// ---------------------------------------------------- soft-DTW forward ----
// Anti-diagonal wavefront; 3 rotating LDS diagonals; full R kept in global
// (needed by backward). R(i,j) for padded i,j in [1,512]; borders only in LDS.
__global__ __launch_bounds__(512) void sdtw_forward_kernel(const float* __restrict__ Dm,
                                                           float* __restrict__ Rg) {
  __shared__ float bufA[NPAD], bufB[NPAD], bufC[NPAD];
  const int b   = blockIdx.x;
  const int tid = threadIdx.x;
  const float* Db = Dm + (size_t)b * NLEN * NLEN;
  float*       Rb = Rg + (size_t)b * NPAD * NPAD;

  for (int k = tid; k < NPAD; k += 512) {
    bufA[k] = (k == 0) ? 0.0f : BIGV;  // diagonal d=0: R(0,0)=0, rest BIG
    bufB[k] = BIGV;                    // diagonal d=1: all BIG (borders)
  }
  __syncthreads();

  float* p2 = bufA;   // diagonal d-2
  float* p1 = bufB;   // diagonal d-1
  float* cur = bufC;  // diagonal d
  const int i = tid + 1;               // row 1..512
  for (int d = 2; d <= 2 * NLEN; ++d) {
    const int j = d - i;
    float r = BIGV;
    if (j >= 1 && j <= NLEN) {
      const float a  = p1[i - 1];      // R(i-1, j)
      const float bb = p2[i - 1];      // R(i-1, j-1)
      const float c  = p1[i];          // R(i, j-1)
      const float m  = fminf(a, fminf(bb, c));
      const float s  = __expf((m - a) * INV_GAMMA)
                     + __expf((m - bb) * INV_GAMMA)
                     + __expf((m - c) * INV_GAMMA);
      r = Db[(i - 1) * NLEN + (j - 1)] + (m - GAMMA_C * __logf(s));
      Rb[i * NPAD + j] = r;
    }
    cur[i] = r;
    if (tid == 0) cur[0] = BIGV;       // border R(0, d)
    __syncthreads();
    float* tmp = p2; p2 = p1; p1 = cur; cur = tmp;
  }
}

// --------------------------------------------------- soft-DTW backward ----
// Reverse wavefront for the alignment matrix E (grad of value wrt D).
// Only two future E-diagonals are needed -> kept in LDS; E*(i-j)^2 is
// accumulated on the fly (E never stored). R/D stay hot in the 192MB L2.
__global__ __launch_bounds__(512) void sdtw_backward_kernel(const float* __restrict__ Dm,
                                                            const float* __restrict__ Rg,
                                                            float* __restrict__ tp) {
  __shared__ float eA[NPAD + 1], eB[NPAD + 1], eC[NPAD + 1]; // i = 0..513
  __shared__ float red[512];
  const int b   = blockIdx.x;
  const int tid = threadIdx.x;
  const float* Db = Dm + (size_t)b * NLEN * NLEN;
  const float* Rb = Rg + (size_t)b * NPAD * NPAD;

  for (int k = tid; k < NPAD + 1; k += 512) { eA[k] = 0.0f; eB[k] = 0.0f; eC[k] = 0.0f; }
  __syncthreads();

  float* ep2 = eA;   // diagonal d+2
  float* ep1 = eB;   // diagonal d+1
  float* cur = eC;   // diagonal d
  const int i = tid + 1;
  float acc = 0.0f;
  for (int d = 2 * NLEN; d >= 2; --d) {
    const int j = d - i;
    float e = 0.0f;
    if (j >= 1 && j <= NLEN) {
      if (i == NLEN && j == NLEN) {
        e = 1.0f;                                  // seed E(N,N)=1
      } else {
        const float rij = Rb[i * NPAD + j];
        if (i + 1 <= NLEN) {                       // child (i+1, j)
          const float t = Rb[(i + 1) * NPAD + j] - rij - Db[i * NLEN + (j - 1)];
          e += __expf(fminf(t * INV_GAMMA, 0.0f)) * ep1[i + 1];
        }
        if (j + 1 <= NLEN) {                       // child (i, j+1)
          const float t = Rb[i * NPAD + (j + 1)] - rij - Db[(i - 1) * NLEN + j];
          e += __expf(fminf(t * INV_GAMMA, 0.0f)) * ep1[i];
        }
        if (i + 1 <= NLEN && j + 1 <= NLEN) {      // child (i+1, j+1)
          const float t = Rb[(i + 1) * NPAD + (j + 1)] - rij - Db[i * NLEN + j];
          e += __expf(fminf(t * INV_GAMMA, 0.0f)) * ep2[i + 1];
        }
      }
      const float dif = (float)(i - j);
      acc += e * dif * dif;                        // fused Omega reduction
    }
    cur[i] = e;                                    // 0 for invalid cells
    __syncthreads();
    float* tmp = ep2; ep2 = ep1; ep1 = cur; cur = tmp;
  }

  red[tid] = acc;
  __syncthreads();
  for (int sft = 256; sft > 0; sft >>= 1) {
    if (tid < sft) red[tid] += red[tid + sft];
    __syncthreads();
  }
  if (tid == 0) tp[b] = red[0];
}

// ------------------------------------------------------------ finalize ----
__global__ void finalize_kernel(const float* __restrict__ Rg,
                                const float* __restrict__ tp,
                                float* __restrict__ out) {
  __shared__ float sv[NBATCH], st[NBATCH];
  const int tid = threadIdx.x;
  if (tid < NBATCH) {
    sv[tid] = Rg[(size_t)tid * NPAD * NPAD + (size_t)NLEN * NPAD + NLEN]; // R(N,N)
    st[tid] = tp[tid];
  }
  __syncthreads();
  if (tid == 0) {
    float a = 0.0f, bsum = 0.0f;
    for (int k = 0; k < NBATCH; ++k) { a += sv[k]; bsum += st[k]; }
    const float loss_shape    = a / (float)NBATCH;
    const float loss_temporal = (bsum / (float)NBATCH) / ((float)NLEN * (float)NLEN);
    out[0] = 0.5f * loss_shape + 0.5f * loss_temporal;
  }
}

// ------------------------------------------------------------- launch -----
extern "C" void kernel_launch(void* const* d_in, const int* in_sizes, int n_in,
                              void* d_out, int out_size, void* d_ws, size_t ws_size,
                              hipStream_t stream) {
  (void)in_sizes; (void)n_in; (void)out_size; (void)ws_size;
  const float* inp = (const float*)d_in[0];   // input  [B,L,DIM] f32
  const float* tgt = (const float*)d_in[1];   // target [B,L,DIM] f32
  float* out = (float*)d_out;
  float* ws  = (float*)d_ws;

  // workspace layout (floats); total ~17.1M floats (~69 MB)
  const size_t off_x  = 0;
  const size_t off_tn = off_x  + (size_t)NBATCH * NLEN * NDIM;   // x buffer
  const size_t off_xn = off_tn + (size_t)NBATCH * NLEN;          // tgt norms
  const size_t off_D  = off_xn + (size_t)NBATCH * NLEN;          // x norms
  const size_t off_R  = off_D  + (size_t)NBATCH * NLEN * NLEN;   // cost matrix
  const size_t off_tp = off_R  + (size_t)NBATCH * NPAD * NPAD;   // DP table

  float* xb = ws + off_x;
  float* tn = ws + off_tn;
  float* xn = ws + off_xn;
  float* Dm = ws + off_D;
  float* Rg = ws + off_R;
  float* tp = ws + off_tp;

  prep_kernel<<<(NBATCH * NLEN + 255) / 256, 256, 0, stream>>>(inp, tgt, xb, tn, xn);

  const int nWaves = NBATCH * (NLEN / 16) * (NLEN / 16);   // 32768 tiles
  dist_wmma_kernel<<<(nWaves * 32) / 256, 256, 0, stream>>>(tgt, xb, tn, xn, Dm);

  sdtw_forward_kernel<<<NBATCH, 512, 0, stream>>>(Dm, Rg);
  sdtw_backward_kernel<<<NBATCH, 512, 0, stream>>>(Dm, Rg, tp);
  finalize_kernel<<<1, 64, 0, stream>>>(Rg, tp, out);
}